// QuantumLayer_22849226015054
// MI455X (gfx1250) — compile-verified
//
#include <hip/hip_runtime.h>

// CDNA5 / gfx1250: wave32, WMMA f32 16x16x4.
typedef float v2f __attribute__((ext_vector_type(2)));
typedef float v8f __attribute__((ext_vector_type(8)));

#define WMMA_F32X4(A, B, C) \
  __builtin_amdgcn_wmma_f32_16x16x4_f32(false, (A), false, (B), (short)0, (C), false, false)

#define NQ 4

// ---------------------------------------------------------------------------
// Phase 1: build the batch-independent 16x16 complex unitary per layer:
//   G_L = CNOT(3,0)*CNOT(0,1)*CNOT(1,2)*CNOT(2,3) * (RZ tensor) * (RY tensor)
// 64 threads = (layer 0..3) x (basis column 0..15). Each thread pushes one
// basis vector through the gates and writes its column.
// ws layout: float G[4][2][16][16]   (layer, re/im, row m, col k)
// ---------------------------------------------------------------------------
__global__ void build_layer_unitaries(const float* __restrict__ w,
                                      float* __restrict__ G) {
  const int tid = threadIdx.x;
  if (tid >= 64) return;
  const int layer = tid >> 4;
  const int col   = tid & 15;
  const float* p  = w + 8 * layer;

  float sr[16], si[16];
#pragma unroll
  for (int i = 0; i < 16; ++i) { sr[i] = 0.f; si[i] = 0.f; }
  sr[col] = 1.f;

  // RY(p[q]) on qubit q. Qubit q lives at bit (3-q) of the flat index.
#pragma unroll
  for (int q = 0; q < NQ; ++q) {
    const float c = cosf(0.5f * p[q]);
    const float s = sinf(0.5f * p[q]);
    const int st = 1 << (3 - q);
#pragma unroll
    for (int m = 0; m < 16; ++m) {
      if (m & st) continue;
      const float ar = sr[m],    ai = si[m];
      const float br = sr[m+st], bi = si[m+st];
      sr[m]    = c*ar - s*br;  si[m]    = c*ai - s*bi;
      sr[m+st] = s*ar + c*br;  si[m+st] = s*ai + c*bi;
    }
  }
  // RZ(p[q+4]) on qubit q: diag(e^{-i a/2}, e^{+i a/2})
#pragma unroll
  for (int q = 0; q < NQ; ++q) {
    const float cz = cosf(0.5f * p[q + NQ]);
    const float sz = sinf(0.5f * p[q + NQ]);
    const int st = 1 << (3 - q);
#pragma unroll
    for (int m = 0; m < 16; ++m) {
      if (m & st) continue;
      float ar = sr[m], ai = si[m];
      sr[m] = ar*cz + ai*sz;  si[m] = ai*cz - ar*sz;
      float br = sr[m+st], bi = si[m+st];
      sr[m+st] = br*cz - bi*sz;  si[m+st] = bi*cz + br*sz;
    }
  }
  // CNOT pairs (control, target): (2,3), (1,2), (0,1), (3,0)
  const int cc[4] = {2, 1, 0, 3};
  const int tt[4] = {3, 2, 1, 0};
#pragma unroll
  for (int k = 0; k < 4; ++k) {
    const int cb = 1 << (3 - cc[k]);
    const int tb = 1 << (3 - tt[k]);
#pragma unroll
    for (int m = 0; m < 16; ++m) {
      if ((m & cb) && !(m & tb)) {
        const int m2 = m | tb;
        float tr = sr[m], ti = si[m];
        sr[m] = sr[m2];  si[m] = si[m2];
        sr[m2] = tr;     si[m2] = ti;
      }
    }
  }
  float* Gr = G + layer * 512;
  float* Gi = Gr + 256;
#pragma unroll
  for (int m = 0; m < 16; ++m) {
    Gr[m * 16 + col] = sr[m];
    Gi[m * 16 + col] = si[m];
  }
}

// ---------------------------------------------------------------------------
// Phase 2: one wave simulates 16 batch samples as the columns of a 16x16
// state matrix held in WMMA C/D layout:
//   register g, lane l : row m = g + 8*(l>=16), column n = l % 16.
// Per layer: per-sample RX tensor product (in-lane FMAs for qubits 1..3,
// one ds_bpermute round fused with qubit 0 + the B-operand half-wave image),
// then the complex 16x16x16 matmul G_L * S as 16 chained
// v_wmma_f32_16x16x4_f32 (Dr and Di are two independent accumulator chains).
// ---------------------------------------------------------------------------

// In-lane RX for stride K in {4,2,1}: pairs (g, g^K); RX is symmetric so the
// update is uniform: r' = c*r + s*i_partner ; i' = c*i - s*r_partner.
#define RX_INLANE(K, C, S)                                        \
  do {                                                            \
    v8f _nr, _ni;                                                 \
    _Pragma("unroll")                                             \
    for (int g = 0; g < 8; ++g) {                                 \
      _nr[g] = (C) * Sr[g] + (S) * Si[g ^ (K)];                   \
      _ni[g] = (C) * Si[g] - (S) * Sr[g ^ (K)];                   \
    }                                                             \
    Sr = _nr; Si = _ni;                                           \
  } while (0)

#define SEL(A, B) (hi ? (A) : (B))

__global__ __launch_bounds__(256) void qsim_wmma_kernel(
    const float* __restrict__ x, const float* __restrict__ G,
    float* __restrict__ out, int nbatch) {
  const int lane = threadIdx.x & 31;
  const int m  = lane & 15;   // A-layout row / C-layout column
  const int hi = lane >> 4;   // half-wave select
  const int waves_per_block = blockDim.x >> 5;
  const int gwave  = blockIdx.x * waves_per_block + (threadIdx.x >> 5);
  const int nwaves = gridDim.x * waves_per_block;
  const int ntiles = nbatch >> 4;
  const int nclamp = nbatch - 1;

  // Load all four layer unitaries into A-matrix layout registers once.
  // A 16x4 f32 layout: lanes 0-15: M=lane, K={0,1}; lanes 16-31: M=lane-16, K={2,3}.
  v2f Ar[4][4], Ai[4][4];
#pragma unroll
  for (int L = 0; L < 4; ++L)
#pragma unroll
    for (int kk = 0; kk < 4; ++kk) {
      const float* base = G + L * 512 + m * 16 + 4 * kk + 2 * hi;
      Ar[L][kk] = *(const v2f*)(base);
      Ai[L][kk] = *(const v2f*)(base + 256);
    }

  for (int t = gwave; t < ntiles; t += nwaves) {
    int n = (t << 4) + m;                 // this lane's batch sample
    if (n > nclamp) n = nclamp;           // keep EXEC full for WMMA
    const float4 xv = reinterpret_cast<const float4*>(x)[n];
    const float c0 = cosf(0.5f * xv.x), s0 = sinf(0.5f * xv.x);
    const float c1 = cosf(0.5f * xv.y), s1 = sinf(0.5f * xv.y);
    const float c2 = cosf(0.5f * xv.z), s2 = sinf(0.5f * xv.z);
    const float c3 = cosf(0.5f * xv.w), s3 = sinf(0.5f * xv.w);

    // |0000> in C/D layout: row 0 lives in register 0 of the low half-wave.
    v8f Sr = {}, Si = {};
    Sr[0] = hi ? 0.f : 1.f;

#pragma unroll
    for (int L = 0; L < 4; ++L) {
      // ---- RX on qubits 1..3 first (in-lane; commutes with qubit 0's RX)
      RX_INLANE(4, c1, s1);   // qubit 1, row stride 4
      RX_INLANE(2, c2, s2);   // qubit 2, row stride 2
      RX_INLANE(1, c3, s3);   // qubit 3, row stride 1

      // ---- Single shuffle round: p = half-wave-swapped S.
      v8f pr, pi;
#pragma unroll
      for (int g = 0; g < 8; ++g) {
        pr[g] = __shfl_xor(Sr[g], 16);
        pi[g] = __shfl_xor(Si[g], 16);
      }
      // Qubit-0 RX fused with the B-operand image:
      //   N = RX0(S)            (this lane's rows)
      //   T = shfl_xor(N, 16)   computed arithmetically — lanes l and l^16
      //                         share the same column, hence the same c0,s0.
      v8f Nr, Ni, Tr, Ti;
#pragma unroll
      for (int g = 0; g < 8; ++g) {
        Nr[g] = c0 * Sr[g] + s0 * pi[g];
        Ni[g] = c0 * Si[g] - s0 * pr[g];
        Tr[g] = c0 * pr[g] + s0 * Si[g];
        Ti[g] = c0 * pi[g] - s0 * Sr[g];
      }

      // ---- B operands (4x16 per K-chunk):
      // B vgpr0 holds row 4kk+2*hi, vgpr1 holds row 4kk+2*hi+1, col = lane%16.
      v2f Br[4], Bi[4], nBi[4];
      Br[0][0] = SEL(Tr[2], Nr[0]);  Br[0][1] = SEL(Tr[3], Nr[1]);
      Br[1][0] = SEL(Tr[6], Nr[4]);  Br[1][1] = SEL(Tr[7], Nr[5]);
      Br[2][0] = SEL(Nr[2], Tr[0]);  Br[2][1] = SEL(Nr[3], Tr[1]);
      Br[3][0] = SEL(Nr[6], Tr[4]);  Br[3][1] = SEL(Nr[7], Tr[5]);
      Bi[0][0] = SEL(Ti[2], Ni[0]);  Bi[0][1] = SEL(Ti[3], Ni[1]);
      Bi[1][0] = SEL(Ti[6], Ni[4]);  Bi[1][1] = SEL(Ti[7], Ni[5]);
      Bi[2][0] = SEL(Ni[2], Ti[0]);  Bi[2][1] = SEL(Ni[3], Ti[1]);
      Bi[3][0] = SEL(Ni[6], Ti[4]);  Bi[3][1] = SEL(Ni[7], Ti[5]);
#pragma unroll
      for (int kk = 0; kk < 4; ++kk) nBi[kk] = -Bi[kk];

      // ---- Complex matmul on the matrix pipes:
      //  Dr = Gr*Sr + Gi*(-Si) ; Di = Gr*Si + Gi*Sr   (K=16 as 4 chunks of 4)
      v8f dr = {}, di = {};
#pragma unroll
      for (int kk = 0; kk < 4; ++kk) dr = WMMA_F32X4(Ar[L][kk], Br[kk],  dr);
#pragma unroll
      for (int kk = 0; kk < 4; ++kk) dr = WMMA_F32X4(Ai[L][kk], nBi[kk], dr);
#pragma unroll
      for (int kk = 0; kk < 4; ++kk) di = WMMA_F32X4(Ar[L][kk], Bi[kk],  di);
#pragma unroll
      for (int kk = 0; kk < 4; ++kk) di = WMMA_F32X4(Ai[L][kk], Br[kk],  di);
      Sr = dr; Si = di;
    }

    // ---- Measurement: out_i = sum_m sign_i(m) * |S[m,n]|^2, sign=+ if bit(3-i)=0.
    float tot = 0.f, ob1 = 0.f, ob2 = 0.f, ob3 = 0.f;
#pragma unroll
    for (int g = 0; g < 8; ++g) {
      const float pg = Sr[g] * Sr[g] + Si[g] * Si[g];
      tot += pg;                                   // qubit 0 sign from hi below
      ob1 += ((g >> 2) & 1) ? -pg : pg;            // qubit 1: bit 2 of m
      ob2 += ((g >> 1) & 1) ? -pg : pg;            // qubit 2: bit 1 of m
      ob3 += ((g >> 0) & 1) ? -pg : pg;            // qubit 3: bit 0 of m
    }
    float ob0 = hi ? -tot : tot;                   // qubit 0: bit 3 of m = hi
    ob0 += __shfl_xor(ob0, 16);
    ob1 += __shfl_xor(ob1, 16);
    ob2 += __shfl_xor(ob2, 16);
    ob3 += __shfl_xor(ob3, 16);

    if (hi == 0) {                                 // lanes 0..15 own columns
      float4 o; o.x = ob0; o.y = ob1; o.z = ob2; o.w = ob3;
      reinterpret_cast<float4*>(out)[n] = o;
    }
  }
}

// ---------------------------------------------------------------------------
extern "C" void kernel_launch(void* const* d_in, const int* in_sizes, int n_in,
                              void* d_out, int out_size, void* d_ws, size_t ws_size,
                              hipStream_t stream) {
  const float* x = (const float*)d_in[0];      // (B, 4) float32
  const float* w = (const float*)d_in[1];      // (32,)  float32
  float* out = (float*)d_out;                  // (B, 4) float32
  float* G   = (float*)d_ws;                   // needs 4*2*256*4 = 8 KB

  const int nbatch = in_sizes[0] / NQ;
  const int ntiles = nbatch >> 4;

  hipLaunchKernelGGL(build_layer_unitaries, dim3(1), dim3(64), 0, stream, w, G);

  int blocks = 512;                             // 4096 waves -> 8 tiles/wave at B=512K
  const int maxb = (ntiles + 7) / 8;            // don't over-launch on tiny batches
  if (blocks > maxb) blocks = maxb;
  if (blocks < 1) blocks = 1;
  hipLaunchKernelGGL(qsim_wmma_kernel, dim3(blocks), dim3(256), 0, stream,
                     x, G, out, nbatch);
}